// reward_loss_41257455845840
// MI455X (gfx1250) — compile-verified
//
#include <hip/hip_runtime.h>

typedef float v2f __attribute__((ext_vector_type(2)));
typedef float v4f __attribute__((ext_vector_type(4)));
typedef float v8f __attribute__((ext_vector_type(8)));
typedef int   v2i __attribute__((ext_vector_type(2)));

// Cross-lane (wave32) sum via V_WMMA_F32_16X16X4_F32.
// A-matrix (16x4 f32, 2 VGPRs/lane): VGPR0 holds A[lane][0] (lanes 0-15) and
// A[lane-16][2] (lanes 16-31); VGPR1 (K=1/K=3) is zeroed.
// B is all ones, so D[m][n] = sum_k A[m][k] = v_m + v_{16+m}, independent of
// B's exact VGPR striping. D layout: lanes 0-15 hold rows 0-7 in d[0..7],
// lanes 16-31 hold rows 8-15. Per-lane sum of d + one xor-16 shuffle gives the
// full 32-lane sum in every lane.
__device__ __forceinline__ float wave_sum_wmma(float v) {
    v2f a; a[0] = v;    a[1] = 0.0f;
    v2f b; b[0] = 1.0f; b[1] = 1.0f;
    v8f c = {};
    v8f d = __builtin_amdgcn_wmma_f32_16x16x4_f32(
        /*neg_a=*/false, a, /*neg_b=*/false, b,
        /*c_mod=*/(short)0, c, /*reuse_a=*/false, /*reuse_b=*/false);
    float s = ((d[0] + d[1]) + (d[2] + d[3])) + ((d[4] + d[5]) + (d[6] + d[7]));
    s += __shfl_xor(s, 16, 32);
    return s;
}

__device__ __forceinline__ void block_reduce_pair(float a1, float a2,
                                                  float* r1, float* r2) {
    float s1 = wave_sum_wmma(a1);
    float s2 = wave_sum_wmma(a2);
    __shared__ float ls1[8], ls2[8];
    const int lane = threadIdx.x & 31;
    const int wave = threadIdx.x >> 5;
    if (lane == 0) { ls1[wave] = s1; ls2[wave] = s2; }
    __syncthreads();
    float b1 = 0.0f, b2 = 0.0f;
    if (threadIdx.x == 0) {
        #pragma unroll
        for (int w = 0; w < 8; ++w) { b1 += ls1[w]; b2 += ls2[w]; }
    }
    *r1 = b1; *r2 = b2;
}

// Stage 1: streamed, coalesced, non-temporal pass over all rows, two rows per
// iteration so every access is GLOBAL_LOAD_B128 / B64 (512 B per wave32 per
// B128). Each block writes one (sum(win*odds*bet), sum(bet)) partial to d_ws.
__global__ __launch_bounds__(256) void reward_partials_kernel(
    const v4f* __restrict__ out4,   // outputs, (N,2) f32, 2 rows per v4f
    const v2i* __restrict__ tgt2,   // targets, (N,) int32, 2 per v2i
    const v4f* __restrict__ in4,    // inputs,  (N,2) f32, 2 rows per v4f
    float2* __restrict__ partials,  // one float2 per block
    int n) {
    const int tid    = blockIdx.x * blockDim.x + threadIdx.x;
    const int stride = gridDim.x * blockDim.x;
    const int npairs = n >> 1;

    float a1 = 0.0f;  // sum of win * odds * bet
    float a2 = 0.0f;  // sum of bet
    for (int j = tid; j < npairs; j += stride) {
        v4f o = __builtin_nontemporal_load(out4 + j);
        v4f q = __builtin_nontemporal_load(in4 + j);
        v2i t = __builtin_nontemporal_load(tgt2 + j);

        float bet0  = fabsf(o[0] - o[1]);
        float bet1  = fabsf(o[2] - o[3]);
        float win0  = t[0] ? o[1] : o[0];
        float odds0 = t[0] ? q[1] : q[0];
        float win1  = t[1] ? o[3] : o[2];
        float odds1 = t[1] ? q[3] : q[2];

        a1 = fmaf(win0 * odds0, bet0, a1);
        a1 = fmaf(win1 * odds1, bet1, a1);
        a2 += bet0 + bet1;
    }

    // Odd-N tail: one thread picks up the last row. Structured divergence:
    // EXEC is restored to all-ones before the WMMA reduction below.
    if ((n & 1) && tid == 0) {
        const float* of = (const float*)out4;
        const float* qf = (const float*)in4;
        const int*   tf = (const int*)tgt2;
        const int i = n - 1;
        float o0 = of[2 * i], o1 = of[2 * i + 1];
        float q0 = qf[2 * i], q1 = qf[2 * i + 1];
        int   t  = tf[i];
        float bet  = fabsf(o0 - o1);
        float win  = t ? o1 : o0;
        float odds = t ? q1 : q0;
        a1 = fmaf(win * odds, bet, a1);
        a2 += bet;
    }

    float b1, b2;
    block_reduce_pair(a1, a2, &b1, &b2);
    if (threadIdx.x == 0) partials[blockIdx.x] = make_float2(b1, b2);
}

// Stage 2: one block folds the per-block partials and emits the scalar loss.
__global__ __launch_bounds__(256) void reward_final_kernel(
    const float2* __restrict__ partials, int nparts,
    float* __restrict__ out, float invN) {
    float a1 = 0.0f, a2 = 0.0f;
    for (int i = threadIdx.x; i < nparts; i += blockDim.x) {
        float2 p = partials[i];
        a1 += p.x;
        a2 += p.y;
    }
    float b1, b2;
    block_reduce_pair(a1, a2, &b1, &b2);
    if (threadIdx.x == 0) {
        // -mean(win*odds*bet) + mean(bet)
        out[0] = (b2 - b1) * invN;
    }
}

extern "C" void kernel_launch(void* const* d_in, const int* in_sizes, int n_in,
                              void* d_out, int out_size, void* d_ws, size_t ws_size,
                              hipStream_t stream) {
    const v4f* outputs = (const v4f*)d_in[0];   // (N,2) float32
    const v2i* targets = (const v2i*)d_in[1];   // (N,)  int32
    const v4f* inputs  = (const v4f*)d_in[2];   // (N,2) float32
    const int  n       = in_sizes[1];           // N rows (targets is length N)

    constexpr int THREADS = 256;   // 8 wave32s per block
    constexpr int BLOCKS  = 2048;  // saturate WGPs; 2048*8B = 16KB of d_ws

    float2* partials = (float2*)d_ws;

    reward_partials_kernel<<<BLOCKS, THREADS, 0, stream>>>(
        outputs, targets, inputs, partials, n);
    reward_final_kernel<<<1, THREADS, 0, stream>>>(
        partials, BLOCKS, (float*)d_out, 1.0f / (float)n);
}